// KCA_75840532513085
// MI455X (gfx1250) — compile-verified
//
#include <hip/hip_runtime.h>
#include <math.h>

// ---------------------------------------------------------------------------
// Problem constants (from the reference)
// ---------------------------------------------------------------------------
#define B_   32
#define N_   5000
#define E_   500
#define ATT_ 50
#define TT_  4
#define IT_  24
#define FT_  384
#define FI_  383
#define REL_ 96           // IT_*TT_
#define H_   100
#define BN_  (B_ * N_)

typedef float v2f __attribute__((ext_vector_type(2)));
typedef float v8f __attribute__((ext_vector_type(8)));

__device__ __forceinline__ float leakyf(float x) { return x > 0.f ? x : 0.1f * x; }
__device__ __forceinline__ float sigm(float x)   { return 1.f / (1.f + __expf(-x)); }

__device__ __forceinline__ void row_stats(const float* p, int nelem, float& mu, float& rv) {
  float m = 0.f;
  for (int i = 0; i < nelem; ++i) m += p[i];
  m /= (float)nelem;
  float v = 0.f;
  for (int i = 0; i < nelem; ++i) { float d = p[i] - m; v += d * d; }
  v /= (float)nelem;
  mu = m;
  rv = rsqrtf(v + 1e-5f);
}

// ---------------------------------------------------------------------------
// fp32 WMMA: D(16x16) = A(16x4) * B(4x16) + C.  Per ISA 7.12.2:
//  A: lane L holds row m=L%16; VGPR0/1 = K = 2*(L/16), 2*(L/16)+1
//  B: lane L holds col n=L%16; VGPR0/1 = K = 2*(L/16), 2*(L/16)+1
//  C/D: VGPR j, lane L -> (M = j + 8*(L/16), N = L%16)
// ---------------------------------------------------------------------------
__device__ __forceinline__ v8f wmma4(v2f a, v2f b, v8f c) {
#if __has_builtin(__builtin_amdgcn_wmma_f32_16x16x4_f32)
  return __builtin_amdgcn_wmma_f32_16x16x4_f32(false, a, false, b, (short)0, c, false, false);
#else
  // shuffle-based emulation (compile-safe fallback; same data layout)
  int lane = threadIdx.x & 31;
  int hh = lane >> 4, cl = lane & 15;
  for (int j = 0; j < 8; ++j) {
    int m = j + 8 * hh;
    float a0 = __shfl(a.x, m, 32),      a1 = __shfl(a.y, m, 32);
    float a2 = __shfl(a.x, m + 16, 32), a3 = __shfl(a.y, m + 16, 32);
    float b0 = __shfl(b.x, cl, 32),      b1 = __shfl(b.y, cl, 32);
    float b2 = __shfl(b.x, cl + 16, 32), b3 = __shfl(b.y, cl + 16, 32);
    c[j] += a0 * b0 + a1 * b1 + a2 * b2 + a3 * b3;
  }
  return c;
#endif
}

// ---------------------------------------------------------------------------
// K0: zero-pad a (srcRows x cols) row-major matrix into (dstRows x cols)
// ---------------------------------------------------------------------------
__global__ __launch_bounds__(256) void k_pad(const float* __restrict__ src,
                                             float* __restrict__ dst,
                                             int srcRows, int dstRows, int cols) {
  int idx = blockIdx.x * 256 + threadIdx.x;
  int total = dstRows * cols;
  if (idx < total) {
    int r = idx / cols;
    dst[idx] = (r < srcRows) ? src[idx] : 0.f;
  }
}

// ---------------------------------------------------------------------------
// K1: text token encoder   (N*TT rows)  row = leaky(W_text @ x + b_text)
// ---------------------------------------------------------------------------
__global__ __launch_bounds__(64) void k_text_emb(const float* __restrict__ X,
                                                 const float* __restrict__ W,
                                                 const float* __restrict__ bvec,
                                                 float* __restrict__ outw) {
  int row = blockIdx.x;                      // n*TT + t
  __shared__ float sx[FT_];
  for (int f = threadIdx.x; f < FT_; f += 64) sx[f] = X[(size_t)row * FT_ + f];
  __syncthreads();
  int d = threadIdx.x;
  if (d < ATT_) {
    const float* w = W + (size_t)d * FT_;
    float acc = bvec[d];
    for (int f = 0; f < FT_; ++f) acc += sx[f] * w[f];
    outw[(size_t)row * ATT_ + d] = leakyf(acc);
  }
}

// ---------------------------------------------------------------------------
// K2: img token encoder   (N*IT rows)
// ---------------------------------------------------------------------------
__global__ __launch_bounds__(64) void k_img_emb(const float* __restrict__ X,
                                                const float* __restrict__ W,
                                                const float* __restrict__ bvec,
                                                float* __restrict__ outw) {
  int row = blockIdx.x;                      // n*IT + i
  __shared__ float sx[FI_];
  for (int f = threadIdx.x; f < FI_; f += 64) sx[f] = X[(size_t)row * FI_ + f];
  __syncthreads();
  int d = threadIdx.x;
  if (d < ATT_) {
    const float* w = W + (size_t)d * FI_;
    float acc = bvec[d];
    for (int f = 0; f < FI_; ++f) acc += sx[f] * w[f];
    outw[(size_t)row * ATT_ + d] = leakyf(acc);
  }
}

// ---------------------------------------------------------------------------
// K3: per-n cross attention: cross, LN(img_att), LN(text_att)
// ---------------------------------------------------------------------------
__global__ __launch_bounds__(128) void k_attn(const float* __restrict__ wsText,
                                              const float* __restrict__ wsImg,
                                              const float* __restrict__ lnG,
                                              const float* __restrict__ lnB,
                                              float* __restrict__ wsCross,
                                              float* __restrict__ wsLnI,
                                              float* __restrict__ wsLnT) {
  int n = blockIdx.x, tid = threadIdx.x;
  __shared__ float te[TT_ * ATT_], ie[IT_ * ATT_];
  __shared__ float cr[REL_], wti[REL_], wit[REL_];
  __shared__ float buf[IT_ * ATT_];
  __shared__ float mt[TT_], zt[TT_], mu[IT_], rv[IT_];

  for (int k = tid; k < TT_ * ATT_; k += 128) te[k] = wsText[(size_t)n * TT_ * ATT_ + k];
  for (int k = tid; k < IT_ * ATT_; k += 128) ie[k] = wsImg[(size_t)n * IT_ * ATT_ + k];
  __syncthreads();
  // cross[i][t] = img_emb[i] . text_emb[t]
  if (tid < REL_) {
    int i = tid / TT_, t = tid % TT_;
    float s = 0.f;
    for (int d = 0; d < ATT_; ++d) s += ie[i * ATT_ + d] * te[t * ATT_ + d];
    cr[tid] = s;
    wsCross[(size_t)n * REL_ + tid] = s;
  }
  __syncthreads();
  // img_att: softmax over i (per t)
  if (tid < TT_) {
    float m = -1e30f;
    for (int i = 0; i < IT_; ++i) m = fmaxf(m, cr[i * TT_ + tid]);
    float z = 0.f;
    for (int i = 0; i < IT_; ++i) z += __expf(cr[i * TT_ + tid] - m);
    mt[tid] = m; zt[tid] = z;
  }
  __syncthreads();
  if (tid < REL_) {
    int t = tid / IT_, i = tid % IT_;
    wti[tid] = __expf(cr[i * TT_ + t] - mt[t]) / zt[t];
  }
  __syncthreads();
  if (tid < TT_ * ATT_ / 2) {            // 100 threads, 2 elems each
    int t = tid / ATT_ * 2, d = tid % ATT_;
    for (int tt = t; tt < t + 2; ++tt) {
      float s = 0.f;
      for (int i = 0; i < IT_; ++i) s += wti[tt * IT_ + i] * ie[i * ATT_ + d];
      buf[tt * ATT_ + d] = s;
    }
  }
  __syncthreads();
  if (tid < TT_) row_stats(&buf[tid * ATT_], ATT_, mu[tid], rv[tid]);
  __syncthreads();
  if (tid < TT_ * ATT_ / 2) {
    int t = tid / ATT_ * 2, d = tid % ATT_;
    for (int tt = t; tt < t + 2; ++tt)
      wsLnI[(size_t)n * TT_ * ATT_ + tt * ATT_ + d] =
          (buf[tt * ATT_ + d] - mu[tt]) * rv[tt] * lnG[d] + lnB[d];
  }
  __syncthreads();
  // text_att: softmax over t (per i)
  if (tid < IT_) {
    float m = -1e30f;
    for (int t = 0; t < TT_; ++t) m = fmaxf(m, cr[tid * TT_ + t]);
    float z = 0.f;
    for (int t = 0; t < TT_; ++t) z += __expf(cr[tid * TT_ + t] - m);
    for (int t = 0; t < TT_; ++t) wit[tid * TT_ + t] = __expf(cr[tid * TT_ + t] - m) / z;
  }
  __syncthreads();
  if (tid < ATT_) {                      // column-per-thread, div-free inner loop
    for (int i = 0; i < IT_; ++i) {
      float s = 0.f;
      for (int t = 0; t < TT_; ++t) s += wit[i * TT_ + t] * te[t * ATT_ + tid];
      buf[i * ATT_ + tid] = s;
    }
  }
  __syncthreads();
  if (tid < IT_) row_stats(&buf[tid * ATT_], ATT_, mu[tid], rv[tid]);
  __syncthreads();
  if (tid < ATT_) {
    float g = lnG[tid], bb = lnB[tid];
    for (int i = 0; i < IT_; ++i)
      wsLnT[(size_t)n * IT_ * ATT_ + i * ATT_ + tid] =
          (buf[i * ATT_ + tid] - mu[i]) * rv[i] * g + bb;
  }
}

// ---------------------------------------------------------------------------
// K4: entW1a[n][j] = entity[n] . W1[j, :E]
// ---------------------------------------------------------------------------
__global__ __launch_bounds__(128) void k_entW1a(const float* __restrict__ ent,
                                                const float* __restrict__ W1,
                                                float* __restrict__ outw) {
  int n = blockIdx.x, tid = threadIdx.x;
  __shared__ float se[E_];
  for (int k = tid; k < E_; k += 128) se[k] = ent[(size_t)n * E_ + k];
  __syncthreads();
  if (tid < H_) {
    const float* w = W1 + (size_t)tid * (2 * E_);
    float acc = 0.f;
    for (int e = 0; e < E_; ++e) acc += se[e] * w[e];
    outw[(size_t)n * H_ + tid] = acc;
  }
}

// ---------------------------------------------------------------------------
// K5: per-b relation gates rg_i, rg_t and rel@W1b
// ---------------------------------------------------------------------------
__global__ __launch_bounds__(128) void k_perb(const int* __restrict__ ps,
                                              const float* __restrict__ rele,
                                              const float* __restrict__ Wr1,
                                              const float* __restrict__ br1,
                                              const float* __restrict__ Wr2,
                                              const float* __restrict__ br2,
                                              const float* __restrict__ W1,
                                              float* __restrict__ rgI,
                                              float* __restrict__ rgT,
                                              float* __restrict__ rW1b) {
  int b = blockIdx.x, tid = threadIdx.x;
  int ridx = ps[b * 3 + 1];
  __shared__ float sr[E_];
  for (int k = tid; k < E_; k += 128) sr[k] = rele[(size_t)ridx * E_ + k];
  __syncthreads();
  if (tid < REL_) {
    float a1 = br1[tid], a2 = br2[tid];
    const float* w1r = Wr1 + (size_t)tid * E_;
    const float* w2r = Wr2 + (size_t)tid * E_;
    for (int e = 0; e < E_; ++e) { a1 += sr[e] * w1r[e]; a2 += sr[e] * w2r[e]; }
    rgI[b * REL_ + tid] = sigm(a1);
    rgT[b * REL_ + tid] = sigm(a2);
  }
  if (tid < H_) {
    const float* w = W1 + (size_t)tid * (2 * E_) + E_;
    float acc = 0.f;
    for (int e = 0; e < E_; ++e) acc += sr[e] * w[e];
    rW1b[b * H_ + tid] = acc;
  }
}

// ---------------------------------------------------------------------------
// K6: per-b positive-sample vectors + norms (32 scalar blocks; cheap)
// ---------------------------------------------------------------------------
__global__ __launch_bounds__(256) void k_pos(const int* __restrict__ ps,
                                             const float* __restrict__ ent,
                                             const float* __restrict__ wsText,
                                             const float* __restrict__ wsImg,
                                             const float* __restrict__ wsCross,
                                             const float* __restrict__ wsLnI,
                                             const float* __restrict__ wsLnT,
                                             const float* __restrict__ rgIbuf,
                                             const float* __restrict__ rgTbuf,
                                             const float* __restrict__ ew1a,
                                             const float* __restrict__ rw1b,
                                             const float* __restrict__ Wip,
                                             const float* __restrict__ bip,
                                             const float* __restrict__ Wtp,
                                             const float* __restrict__ btp,
                                             const float* __restrict__ b1,
                                             const float* __restrict__ W3,
                                             const float* __restrict__ b3,
                                             const float* __restrict__ lnG,
                                             const float* __restrict__ lnB,
                                             float* __restrict__ pImg,
                                             float* __restrict__ pTxt,
                                             float* __restrict__ pT,
                                             float* __restrict__ pNrm) {
  int b = blockIdx.x, tid = threadIdx.x;
  int n = ps[b * 3 + 2];
  __shared__ float cr[REL_], ie[IT_ * ATT_], te[TT_ * ATT_];
  __shared__ float lnI[TT_ * ATT_], lnT[IT_ * ATT_], rgI[REL_], rgT[REL_];
  __shared__ float tile[IT_ * ATT_], h[H_];
  __shared__ float mu[IT_], rv[IT_], acc3[3];
  if (tid < 3) acc3[tid] = 0.f;
  for (int k = tid; k < REL_; k += 256) {
    cr[k] = wsCross[(size_t)n * REL_ + k];
    rgI[k] = rgIbuf[b * REL_ + k];
    rgT[k] = rgTbuf[b * REL_ + k];
  }
  for (int k = tid; k < IT_ * ATT_; k += 256) {
    ie[k] = wsImg[(size_t)n * IT_ * ATT_ + k];
    lnT[k] = wsLnT[(size_t)n * IT_ * ATT_ + k];
  }
  for (int k = tid; k < TT_ * ATT_; k += 256) {
    te[k] = wsText[(size_t)n * TT_ * ATT_ + k];
    lnI[k] = wsLnI[(size_t)n * TT_ * ATT_ + k];
  }
  __syncthreads();
  // ---- img path
  if (tid < TT_ * ATT_) {
    int t = tid / ATT_, d = tid % ATT_;
    float v = 0.f;
    for (int i = 0; i < IT_; ++i) v += rgI[i * TT_ + t] * cr[i * TT_ + t] * ie[i * ATT_ + d];
    tile[tid] = v;
  }
  __syncthreads();
  if (tid < TT_) row_stats(&tile[tid * ATT_], ATT_, mu[tid], rv[tid]);
  __syncthreads();
  if (tid < TT_ * ATT_) {
    int t = tid / ATT_, d = tid % ATT_;
    tile[tid] = (tile[tid] - mu[t]) * rv[t] * lnG[d] + lnB[d] + lnI[tid];
  }
  __syncthreads();
  if (tid < ATT_) {
    const float* w = Wip + (size_t)tid * (TT_ * ATT_);
    float v = bip[tid];
    for (int k = 0; k < TT_ * ATT_; ++k) v += tile[k] * w[k];
    pImg[b * ATT_ + tid] = v;
    atomicAdd(&acc3[0], v * v);
  }
  __syncthreads();
  // ---- text path
  if (tid < 200) {                       // 200 threads, 6 rows each
    int d = tid % ATT_, i0 = (tid / ATT_) * 6;
    for (int i = i0; i < i0 + 6; ++i) {
      float v = 0.f;
      for (int t = 0; t < TT_; ++t) v += rgT[i * TT_ + t] * cr[i * TT_ + t] * te[t * ATT_ + d];
      tile[i * ATT_ + d] = v;
    }
  }
  __syncthreads();
  if (tid < IT_) row_stats(&tile[tid * ATT_], ATT_, mu[tid], rv[tid]);
  __syncthreads();
  if (tid < 200) {
    int d = tid % ATT_, i0 = (tid / ATT_) * 6;
    float g = lnG[d], bb = lnB[d];
    for (int i = i0; i < i0 + 6; ++i)
      tile[i * ATT_ + d] = (tile[i * ATT_ + d] - mu[i]) * rv[i] * g + bb + lnT[i * ATT_ + d];
  }
  __syncthreads();
  if (tid < ATT_) {
    const float* w = Wtp + (size_t)tid * (IT_ * ATT_);
    float v = btp[tid];
    for (int k = 0; k < IT_ * ATT_; ++k) v += tile[k] * w[k];
    pTxt[b * ATT_ + tid] = v;
    atomicAdd(&acc3[1], v * v);
  }
  __syncthreads();
  // ---- t path
  if (tid < H_) h[tid] = leakyf(ew1a[(size_t)n * H_ + tid] + rw1b[b * H_ + tid] + b1[tid]);
  __syncthreads();
  for (int e = tid; e < E_; e += 256) {
    const float* w = W3 + (size_t)e * H_;
    float a = b3[e];
    for (int j = 0; j < H_; ++j) a += h[j] * w[j];
    float tv = ent[(size_t)n * E_ + e] * sigm(a);
    pT[b * E_ + e] = tv;
    atomicAdd(&acc3[2], tv * tv);
  }
  __syncthreads();
  if (tid < 3) pNrm[b * 3 + tid] = sqrtf(acc3[tid]);
}

// ---------------------------------------------------------------------------
// K7: main fused kernel (img + text projections via V_WMMA_F32_16X16X4_F32)
//     grid = (ceil(N/16), B), 128 threads = 4 wave32s, one wave per column
//     tile with the full K range (compile-time trip counts), software-
//     pipelined fragment loads (load s+1 before WMMA s -> partial waits).
// ---------------------------------------------------------------------------
__global__ __launch_bounds__(128) void k_proj(const float* __restrict__ wsText,
                                              const float* __restrict__ wsImg,
                                              const float* __restrict__ wsCross,
                                              const float* __restrict__ wsLnI,
                                              const float* __restrict__ wsLnT,
                                              const float* __restrict__ rgIbuf,
                                              const float* __restrict__ rgTbuf,
                                              const float* __restrict__ WipP,   // 64 x 200, zero-padded
                                              const float* __restrict__ bip,
                                              const float* __restrict__ WtpP,   // 64 x 1200, zero-padded
                                              const float* __restrict__ btp,
                                              const float* __restrict__ lnG,
                                              const float* __restrict__ lnB,
                                              const float* __restrict__ pImg,
                                              const float* __restrict__ pTxt,
                                              const float* __restrict__ pNrm,
                                              float* __restrict__ outp) {
  const int b = blockIdx.y;
  const int n0 = blockIdx.x * 16;
  const int tid = threadIdx.x;
  const int ct = tid >> 5;        // wave id == output column tile (0..3)
  const int lane = tid & 31;
  const int hh = lane >> 4;       // K-half within fragment
  const int mc = lane & 15;       // A row / D column (lane%16)
  const int colg = ct * 16 + mc;

  __shared__ float sTe[16 * 200];
  __shared__ float sCr[16 * 96];
  __shared__ float sA[16 * 200];     // img A-tile (t*50+d flattened)
  __shared__ float sCk[16 * 100];    // text A-chunk (2 image tokens per pass)
  __shared__ float sOut[16 * 64];
  __shared__ float sRgI[REL_], sRgT[REL_];
  __shared__ float sPI[ATT_], sPT[ATT_];
  __shared__ float sMu[64], sRv[64];
  __shared__ float sPN[2];

  const float* wipRow = WipP + (size_t)colg * 200;
  const float* wtpRow = WtpP + (size_t)colg * 1200;
  __builtin_prefetch(wipRow, 0, 1);               // global_prefetch_b8 -> warm L2
  __builtin_prefetch(wtpRow, 0, 1);

  if (tid < REL_) { sRgI[tid] = rgIbuf[b * REL_ + tid]; sRgT[tid] = rgTbuf[b * REL_ + tid]; }
  if (tid < ATT_) { sPI[tid] = pImg[b * ATT_ + tid]; sPT[tid] = pTxt[b * ATT_ + tid]; }
  if (tid < 2) sPN[tid] = pNrm[b * 3 + tid];
  if (tid < 96) {
    for (int r = 0; r < 16; ++r) {
      int n = n0 + r; if (n >= N_) n = N_ - 1;
      sCr[r * 96 + tid] = wsCross[(size_t)n * 96 + tid];
    }
  }
  for (int c = tid; c < 200; c += 128) {
    for (int r = 0; r < 16; ++r) {
      int n = n0 + r; if (n >= N_) n = N_ - 1;
      sTe[r * 200 + c] = wsText[(size_t)n * 200 + c];
    }
  }
  __syncthreads();

  // ---- img A-tile: rel-gated cross @ img_emb, LayerNorm, + LN(img_att)
  for (int c = tid; c < 200; c += 128) {
    int t = c / ATT_, d = c % ATT_;
    for (int r = 0; r < 16; ++r) {
      int n = n0 + r; if (n >= N_) n = N_ - 1;
      const float* ie = wsImg + (size_t)n * (IT_ * ATT_) + d;
      float v = 0.f;
      for (int i = 0; i < IT_; ++i) v += sRgI[i * 4 + t] * sCr[r * 96 + i * 4 + t] * ie[i * ATT_];
      sA[r * 200 + c] = v;
    }
  }
  __syncthreads();
  if (tid < 64) {
    int r = tid >> 2, t = tid & 3;
    row_stats(&sA[r * 200 + t * ATT_], ATT_, sMu[tid], sRv[tid]);
  }
  __syncthreads();
  for (int c = tid; c < 200; c += 128) {
    int t = c / ATT_, d = c % ATT_;
    float g = lnG[d], bb = lnB[d];
    for (int r = 0; r < 16; ++r) {
      int row = r * 4 + t;
      int n = n0 + r; if (n >= N_) n = N_ - 1;
      sA[r * 200 + c] = (sA[r * 200 + c] - sMu[row]) * sRv[row] * g + bb + wsLnI[(size_t)n * 200 + c];
    }
  }
  __syncthreads();

  // ---- img projection: (16x200)x(200x64pad), 50 K-steps, pipelined
  const float* aRowI = &sA[mc * 200];
  v8f accI = {0.f, 0.f, 0.f, 0.f, 0.f, 0.f, 0.f, 0.f};
  {
    v2f a0 = *(const v2f*)(aRowI + 2 * hh);
    v2f b0 = *(const v2f*)(wipRow + 2 * hh);
    #pragma unroll 7
    for (int s = 1; s < 50; ++s) {
      int kk = s * 4 + 2 * hh;
      v2f a1 = *(const v2f*)(aRowI + kk);
      v2f b1 = *(const v2f*)(wipRow + kk);
      accI = wmma4(a0, b0, accI);
      a0 = a1; b0 = b1;
    }
    accI = wmma4(a0, b0, accI);
  }

  // ---- text projection: K=1200 streamed in 12 chunks of 100 (2 tokens each)
  const float* aRowC = &sCk[mc * 100];
  v8f accT = {0.f, 0.f, 0.f, 0.f, 0.f, 0.f, 0.f, 0.f};
  for (int g2 = 0; g2 < 12; ++g2) {
    __syncthreads();   // previous chunk fully consumed by WMMA
    if (tid < 100) {
      int il = tid / ATT_, d = tid % ATT_, i = 2 * g2 + il;
      float w0 = sRgT[i * 4 + 0], w1 = sRgT[i * 4 + 1], w2 = sRgT[i * 4 + 2], w3 = sRgT[i * 4 + 3];
      for (int r = 0; r < 16; ++r) {
        const float* crr = &sCr[r * 96 + i * 4];
        const float* ter = &sTe[r * 200 + d];
        float v = w0 * crr[0] * ter[0] + w1 * crr[1] * ter[ATT_] +
                  w2 * crr[2] * ter[2 * ATT_] + w3 * crr[3] * ter[3 * ATT_];
        sCk[r * 100 + tid] = v;
      }
    }
    __syncthreads();
    if (tid < 32) {
      int r = tid >> 1, il = tid & 1;
      row_stats(&sCk[r * 100 + il * ATT_], ATT_, sMu[tid], sRv[tid]);
    }
    __syncthreads();
    if (tid < 100) {
      int il = tid / ATT_, d = tid % ATT_, i = 2 * g2 + il;
      float g = lnG[d], bb = lnB[d];
      for (int r = 0; r < 16; ++r) {
        int row = r * 2 + il;
        int n = n0 + r; if (n >= N_) n = N_ - 1;
        sCk[r * 100 + tid] = (sCk[r * 100 + tid] - sMu[row]) * sRv[row] * g + bb +
                             wsLnT[(size_t)n * 1200 + i * ATT_ + d];
      }
    }
    __syncthreads();
    const float* wRow = wtpRow + g2 * 100;
    v2f a0 = *(const v2f*)(aRowC + 2 * hh);
    v2f b0 = *(const v2f*)(wRow + 2 * hh);
    #pragma unroll 4
    for (int s = 1; s < 25; ++s) {
      int kk = s * 4 + 2 * hh;
      v2f a1 = *(const v2f*)(aRowC + kk);
      v2f b1 = *(const v2f*)(wRow + kk);
      accT = wmma4(a0, b0, accT);
      a0 = a1; b0 = b1;
    }
    accT = wmma4(a0, b0, accT);
  }

  // ---- emit simil_img (each D tile has a unique writer wave: plain stores)
  __syncthreads();
  for (int j = 0; j < 8; ++j) sOut[(j + 8 * hh) * 64 + ct * 16 + mc] = accI[j];
  __syncthreads();
  if (tid < 16) {
    int n = n0 + tid;
    if (n < N_) {
      float dp = 0.f, nr = 0.f;
      for (int d = 0; d < ATT_; ++d) {
        float x = sOut[tid * 64 + d] + bip[d];
        dp += x * sPI[d]; nr += x * x;
      }
      outp[BN_ + (size_t)b * N_ + n] = dp / (sqrtf(nr) * sPN[0] + 1e-10f);
    }
  }
  __syncthreads();
  // ---- emit simil_text
  for (int j = 0; j < 8; ++j) sOut[(j + 8 * hh) * 64 + ct * 16 + mc] = accT[j];
  __syncthreads();
  if (tid < 16) {
    int n = n0 + tid;
    if (n < N_) {
      float dp = 0.f, nr = 0.f;
      for (int d = 0; d < ATT_; ++d) {
        float x = sOut[tid * 64 + d] + btp[d];
        dp += x * sPT[d]; nr += x * x;
      }
      outp[2 * BN_ + (size_t)b * N_ + n] = dp / (sqrtf(nr) * sPN[1] + 1e-10f);
    }
  }
}

// ---------------------------------------------------------------------------
// K8: gate GEMM + t_att cosine sim via WMMA: (16x100)x(100x512pad)
// ---------------------------------------------------------------------------
__global__ __launch_bounds__(256) void k_tpath(const float* __restrict__ ent,
                                               const float* __restrict__ ew1a,
                                               const float* __restrict__ rw1b,
                                               const float* __restrict__ b1,
                                               const float* __restrict__ W3P,   // 512 x 100, zero-padded
                                               const float* __restrict__ b3,
                                               const float* __restrict__ pT,
                                               const float* __restrict__ pNrm,
                                               float* __restrict__ outp) {
  const int b = blockIdx.y;
  const int n0 = blockIdx.x * 16;
  const int tid = threadIdx.x, wid = tid >> 5, lane = tid & 31;
  const int hh = lane >> 4, mc = lane & 15;
  __shared__ float sH[16 * H_];
  __shared__ float sPt[E_];
  __shared__ float sDot[16], sNrm[16];
  __shared__ float sPN;
  __builtin_prefetch(W3P + (size_t)(wid * 16 + mc) * H_, 0, 1);
  if (tid < H_) {
    float bj = b1[tid], rw = rw1b[b * H_ + tid];
    for (int r = 0; r < 16; ++r) {
      int n = n0 + r; if (n >= N_) n = N_ - 1;
      sH[r * H_ + tid] = leakyf(ew1a[(size_t)n * H_ + tid] + rw + bj);
    }
  }
  for (int k = tid; k < E_; k += 256) sPt[k] = pT[b * E_ + k];
  if (tid < 16) { sDot[tid] = 0.f; sNrm[tid] = 0.f; }
  if (tid == 0) sPN = pNrm[b * 3 + 2];
  __syncthreads();
  const float* aRow = &sH[mc * H_];
  for (int ct = wid; ct < 32; ct += 8) {          // 32 col tiles over E=500 (padded 512)
    int e = ct * 16 + mc;
    const float* wRow = W3P + (size_t)e * H_;
    v8f acc = {0.f, 0.f, 0.f, 0.f, 0.f, 0.f, 0.f, 0.f};
    v2f a0 = *(const v2f*)(aRow + 2 * hh);
    v2f b0 = *(const v2f*)(wRow + 2 * hh);
    #pragma unroll 4
    for (int s = 1; s < 25; ++s) {
      int kk = s * 4 + 2 * hh;
      v2f a1 = *(const v2f*)(aRow + kk);
      v2f b1 = *(const v2f*)(wRow + kk);
      acc = wmma4(a0, b0, acc);
      a0 = a1; b0 = b1;
    }
    acc = wmma4(a0, b0, acc);
    if (e < E_) {
      float b3e = b3[e], pte = sPt[e];
      for (int j = 0; j < 8; ++j) {
        int m = j + 8 * hh;
        int n = n0 + m; if (n >= N_) n = N_ - 1;
        float tv = ent[(size_t)n * E_ + e] * sigm(acc[j] + b3e);
        atomicAdd(&sDot[m], tv * pte);
        atomicAdd(&sNrm[m], tv * tv);
      }
    }
  }
  __syncthreads();
  if (tid < 16) {
    int n = n0 + tid;
    if (n < N_) outp[3 * BN_ + (size_t)b * N_ + n] = sDot[tid] / (sqrtf(sNrm[tid]) * sPN + 1e-10f);
  }
}

// ---------------------------------------------------------------------------
// K9: out = softmax(simil_t*2) + softmax(simil_img*2) + softmax(simil_text*2)
// ---------------------------------------------------------------------------
__global__ __launch_bounds__(256) void k_softmax(float* __restrict__ outp) {
  const int b = blockIdx.x, tid = threadIdx.x;
  __shared__ float red[256];
  __shared__ float sM, sZ;
  const int secs[3] = {3, 1, 2};   // t first (writes), then img/text (adds)
  for (int a = 0; a < 3; ++a) {
    const float* x = outp + (size_t)secs[a] * BN_ + (size_t)b * N_;
    float mx = -1e30f;
    for (int n = tid; n < N_; n += 256) mx = fmaxf(mx, x[n]);
    red[tid] = mx; __syncthreads();
    for (int s = 128; s > 0; s >>= 1) { if (tid < s) red[tid] = fmaxf(red[tid], red[tid + s]); __syncthreads(); }
    if (tid == 0) sM = red[0];
    __syncthreads();
    float sm = 0.f;
    for (int n = tid; n < N_; n += 256) sm += __expf(2.f * (x[n] - sM));
    red[tid] = sm; __syncthreads();
    for (int s = 128; s > 0; s >>= 1) { if (tid < s) red[tid] += red[tid + s]; __syncthreads(); }
    if (tid == 0) sZ = red[0];
    __syncthreads();
    float* o = outp + (size_t)b * N_;
    for (int n = tid; n < N_; n += 256) {
      float v = __expf(2.f * (x[n] - sM)) / sZ;
      o[n] = (a == 0) ? v : (o[n] + v);
    }
    __syncthreads();
  }
}

// ---------------------------------------------------------------------------
// launch
// ---------------------------------------------------------------------------
extern "C" void kernel_launch(void* const* d_in, const int* in_sizes, int n_in,
                              void* d_out_v, int out_size, void* d_ws, size_t ws_size,
                              hipStream_t stream) {
  (void)in_sizes; (void)n_in; (void)out_size; (void)ws_size;
  const int*   ps   = (const int*)d_in[0];
  const float* tte  = (const float*)d_in[1];
  const float* tie  = (const float*)d_in[2];
  const float* ent  = (const float*)d_in[3];
  const float* rele = (const float*)d_in[4];
  const float* Wt   = (const float*)d_in[5];
  const float* bt   = (const float*)d_in[6];
  const float* Wi   = (const float*)d_in[7];
  const float* bi   = (const float*)d_in[8];
  const float* Wtp  = (const float*)d_in[9];
  const float* btp  = (const float*)d_in[10];
  const float* Wip  = (const float*)d_in[11];
  const float* bip  = (const float*)d_in[12];
  const float* Wr1  = (const float*)d_in[13];
  const float* br1  = (const float*)d_in[14];
  const float* Wr2  = (const float*)d_in[15];
  const float* br2  = (const float*)d_in[16];
  const float* W1   = (const float*)d_in[17];
  const float* b1   = (const float*)d_in[18];
  const float* W3   = (const float*)d_in[19];
  const float* b3   = (const float*)d_in[20];
  const float* lnG  = (const float*)d_in[21];
  const float* lnB  = (const float*)d_in[22];
  float* outp = (float*)d_out_v;
  float* ws   = (float*)d_ws;

  // workspace layout (floats) — ~60.6 MB total, L2-resident on MI455X (192 MB)
  float* wsText = ws;                                 // N*TT*ATT = 1,000,000
  float* wsImg  = wsText + (size_t)N_ * TT_ * ATT_;   // N*IT*ATT = 6,000,000
  float* wsCr   = wsImg  + (size_t)N_ * IT_ * ATT_;   // N*96     =   480,000
  float* wsLnI  = wsCr   + (size_t)N_ * REL_;         //          = 1,000,000
  float* wsLnT  = wsLnI  + (size_t)N_ * TT_ * ATT_;   //          = 6,000,000
  float* wsEW   = wsLnT  + (size_t)N_ * IT_ * ATT_;   // N*100    =   500,000
  float* wsRgI  = wsEW   + (size_t)N_ * H_;           // B*96
  float* wsRgT  = wsRgI  + (size_t)B_ * REL_;
  float* wsRW1b = wsRgT  + (size_t)B_ * REL_;         // B*100
  float* wsPImg = wsRW1b + (size_t)B_ * H_;           // B*50
  float* wsPTxt = wsPImg + (size_t)B_ * ATT_;         // B*50
  float* wsPT   = wsPTxt + (size_t)B_ * ATT_;         // B*500
  float* wsPNrm = wsPT   + (size_t)B_ * E_;           // B*3
  float* wsWipP = wsPNrm + (size_t)B_ * 3;            // 64*200   = 12,800
  float* wsWtpP = wsWipP + (size_t)64 * 200;          // 64*1200  = 76,800
  float* wsW3P  = wsWtpP + (size_t)64 * 1200;         // 512*100  = 51,200

  // pad weights once (branch-free WMMA B-fragment loads)
  k_pad<<<(64 * 200 + 255) / 256, 256, 0, stream>>>(Wip, wsWipP, ATT_, 64, 200);
  k_pad<<<(64 * 1200 + 255) / 256, 256, 0, stream>>>(Wtp, wsWtpP, ATT_, 64, 1200);
  k_pad<<<(512 * 100 + 255) / 256, 256, 0, stream>>>(W3, wsW3P, E_, 512, H_);

  k_text_emb<<<N_ * TT_, 64, 0, stream>>>(tte, Wt, bt, wsText);
  k_img_emb <<<N_ * IT_, 64, 0, stream>>>(tie, Wi, bi, wsImg);
  k_attn    <<<N_, 128, 0, stream>>>(wsText, wsImg, lnG, lnB, wsCr, wsLnI, wsLnT);
  k_entW1a  <<<N_, 128, 0, stream>>>(ent, W1, wsEW);
  k_perb    <<<B_, 128, 0, stream>>>(ps, rele, Wr1, br1, Wr2, br2, W1, wsRgI, wsRgT, wsRW1b);
  k_pos     <<<B_, 256, 0, stream>>>(ps, ent, wsText, wsImg, wsCr, wsLnI, wsLnT,
                                     wsRgI, wsRgT, wsEW, wsRW1b,
                                     Wip, bip, Wtp, btp, b1, W3, b3, lnG, lnB,
                                     wsPImg, wsPTxt, wsPT, wsPNrm);
  dim3 gmain((N_ + 15) / 16, B_);
  k_proj  <<<gmain, 128, 0, stream>>>(wsText, wsImg, wsCr, wsLnI, wsLnT, wsRgI, wsRgT,
                                      wsWipP, bip, wsWtpP, btp, lnG, lnB,
                                      wsPImg, wsPTxt, wsPNrm, outp);
  k_tpath <<<gmain, 256, 0, stream>>>(ent, wsEW, wsRW1b, b1, wsW3P, b3, wsPT, wsPNrm, outp);
  k_softmax<<<B_, 256, 0, stream>>>(outp);
}